// OneDirSAGEConv_83408264888627
// MI455X (gfx1250) — compile-verified
//
#include <hip/hip_runtime.h>

typedef __attribute__((ext_vector_type(2))) float v2f;
typedef __attribute__((ext_vector_type(4))) float v4f;
typedef __attribute__((ext_vector_type(8))) float v8f;

#define D_FEAT 128

// ---------------------------------------------------------------------------
// Kernel 1: zero workspace accumulators (neigh_sum + deg), grid-stride.
// ---------------------------------------------------------------------------
__global__ __launch_bounds__(256) void sage_zero_kernel(float* __restrict__ p, int n) {
  int i = blockIdx.x * 256 + threadIdx.x;
  int stride = gridDim.x * 256;
  for (; i < n; i += stride) p[i] = 0.0f;
}

// ---------------------------------------------------------------------------
// Kernel 2: edge scatter. One wave32 per edge; each lane owns 4 features.
// neigh_sum[dst] += feat[src] * w ; deg[dst] += 1
// ---------------------------------------------------------------------------
__global__ __launch_bounds__(256) void sage_scatter_kernel(
    const float* __restrict__ feat, const int* __restrict__ esrc,
    const int* __restrict__ edst, const float* __restrict__ ew,
    float* __restrict__ nsum, float* __restrict__ deg, int n_edges) {
  unsigned long long gid = (unsigned long long)blockIdx.x * 256ull + threadIdx.x;
  int e = (int)(gid >> 5);
  int lane = (int)(gid & 31);
  if (e >= n_edges) return;
  int s = esrc[e];
  int d = edst[e];
  float w = ew[e];
  v4f v = *((const v4f*)(feat + (size_t)s * D_FEAT) + lane);  // global_load_b128
  float* np = nsum + (size_t)d * D_FEAT + lane * 4;
  atomicAdd(np + 0, v.x * w);
  atomicAdd(np + 1, v.y * w);
  atomicAdd(np + 2, v.z * w);
  atomicAdd(np + 3, v.w * w);
  if (lane == 0) atomicAdd(deg + d, 1.0f);
}

// ---------------------------------------------------------------------------
// Kernel 3: in-place mean: nsum[i] /= max(deg[node], 1)
// ---------------------------------------------------------------------------
__global__ __launch_bounds__(256) void sage_normalize_kernel(
    float* __restrict__ nsum, const float* __restrict__ deg, int total) {
  int i = blockIdx.x * 256 + threadIdx.x;
  if (i >= total) return;
  int node = i >> 7;  // /128
  float dg = deg[node];
  float inv = 1.0f / fmaxf(dg, 1.0f);
  nsum[i] *= inv;
}

// ---------------------------------------------------------------------------
// Kernel 4: fused dual GEMM via V_WMMA_F32_16X16X4_F32.
// out = feat @ Wself^T + hneigh @ Wneigh^T + bias
// One wave computes a 16(M-nodes) x 16(N-outfeat) tile; K looped 4 at a time.
//
// A 16x4 f32 layout : lanes 0-15 hold M=lane {K=0,K=1}; lanes 16-31 {K=2,K=3}
// B 4x16  f32 layout: lanes 0-15 hold N=lane {K=0,K=1}; lanes 16-31 {K=2,K=3}
//   with B[k][n] = W[n][k]  (W row-major [D_OUT][D_IN])
// C/D 16x16 f32     : VGPR v, lanes 0-15 -> M=v, lanes 16-31 -> M=v+8; N=lane&15
// ---------------------------------------------------------------------------
__global__ __launch_bounds__(128) void sage_gemm_kernel(
    const float* __restrict__ feat, const float* __restrict__ hneigh,
    const float* __restrict__ Wself, const float* __restrict__ Wneigh,
    const float* __restrict__ bias, float* __restrict__ out, int n_nodes) {
  const int wave = threadIdx.x >> 5;
  const int lane = threadIdx.x & 31;
  const int tiles_n = D_FEAT / 16;  // 8
  int tile = blockIdx.x * 4 + wave;
  int tm = tile / tiles_n;
  int tn = tile - tm * tiles_n;
  if (tm * 16 >= n_nodes) return;  // wave-uniform: EXEC stays all-ones for WMMA

  const int m0 = tm * 16;
  const int n0 = tn * 16;
  const int lm = lane & 15;
  const int kh = (lane >> 4) << 1;  // 0 for lanes 0-15, 2 for lanes 16-31

  const float* aself = feat   + (size_t)(m0 + lm) * D_FEAT + kh;
  const float* aneigh = hneigh + (size_t)(m0 + lm) * D_FEAT + kh;
  const float* bself = Wself  + (size_t)(n0 + lm) * D_FEAT + kh;
  const float* bneigh = Wneigh + (size_t)(n0 + lm) * D_FEAT + kh;

  v8f acc = {};

#pragma unroll
  for (int k = 0; k < D_FEAT; k += 4) {
    v2f a = *(const v2f*)(aself + k);   // global_load_b64
    v2f b = *(const v2f*)(bself + k);
    acc = __builtin_amdgcn_wmma_f32_16x16x4_f32(
        /*neg_a=*/false, a, /*neg_b=*/false, b,
        /*c_mod=*/(short)0, acc, /*reuse_a=*/false, /*reuse_b=*/false);
  }
#pragma unroll
  for (int k = 0; k < D_FEAT; k += 4) {
    v2f a = *(const v2f*)(aneigh + k);
    v2f b = *(const v2f*)(bneigh + k);
    acc = __builtin_amdgcn_wmma_f32_16x16x4_f32(
        false, a, false, b, (short)0, acc, false, false);
  }

  float bv = bias[n0 + lm];
  int rbase = (lane < 16) ? 0 : 8;
#pragma unroll
  for (int v = 0; v < 8; ++v) {
    out[(size_t)(m0 + rbase + v) * D_FEAT + n0 + lm] = acc[v] + bv;
  }
}

// ---------------------------------------------------------------------------
// Host-side launcher
// ---------------------------------------------------------------------------
extern "C" void kernel_launch(void* const* d_in, const int* in_sizes, int n_in,
                              void* d_out, int out_size, void* d_ws, size_t ws_size,
                              hipStream_t stream) {
  const float* feat = (const float*)d_in[0];
  const int*   esrc = (const int*)d_in[1];
  const int*   edst = (const int*)d_in[2];
  const float* ew   = (const float*)d_in[3];
  const float* Wn   = (const float*)d_in[4];
  const float* Ws   = (const float*)d_in[5];
  const float* bs   = (const float*)d_in[6];
  float* out = (float*)d_out;

  const int n_nodes = in_sizes[0] / D_FEAT;
  const int n_edges = in_sizes[1];

  float* nsum = (float*)d_ws;                      // [n_nodes * 128]
  float* deg  = nsum + (size_t)n_nodes * D_FEAT;   // [n_nodes]

  // 1) zero accumulators
  int ztot = n_nodes * (D_FEAT + 1);
  int zblocks = (ztot + 255) / 256;
  if (zblocks > 4096) zblocks = 4096;
  sage_zero_kernel<<<zblocks, 256, 0, stream>>>(nsum, ztot);

  // 2) edge scatter (wave32 per edge)
  unsigned long long sthreads = (unsigned long long)n_edges * 32ull;
  unsigned int sblocks = (unsigned int)((sthreads + 255ull) / 256ull);
  sage_scatter_kernel<<<sblocks, 256, 0, stream>>>(feat, esrc, edst, ew,
                                                   nsum, deg, n_edges);

  // 3) mean normalize in place
  int ntot = n_nodes * D_FEAT;
  sage_normalize_kernel<<<(ntot + 255) / 256, 256, 0, stream>>>(nsum, deg, ntot);

  // 4) fused dual-GEMM + bias via WMMA (wave per 16x16 tile, 4 waves/block)
  int tiles = ((n_nodes + 15) / 16) * (D_FEAT / 16);
  int gblocks = (tiles + 3) / 4;
  sage_gemm_kernel<<<gblocks, 128, 0, stream>>>(feat, nsum, Ws, Wn, bs, out,
                                                n_nodes);
}